// ComputeEnergyForce_89343909691948
// MI455X (gfx1250) — compile-verified
//
#include <hip/hip_runtime.h>
#include <stdint.h>

// ---------------- problem constants (from reference) ----------------
#define NS      16
#define N_ATOMS 2000
#define NBOND   2000
#define NANG    4000
#define NVDW    400000
#define NTOR    6000
#define NIMP    1000

#define OFF_EBOND   0
#define OFF_EANGLE  (NBOND)                    // 2000
#define OFF_EUB     (NBOND + NANG)             // 6000
#define OFF_EVDW    (OFF_EUB + 1)              // 6001
#define OFF_ECH     (OFF_EVDW + NVDW)          // 406001
#define OFF_ETOR    (OFF_ECH + NVDW)           // 806001
#define OFF_EIMP    (OFF_ETOR + NTOR)          // 812001
#define OFF_FORCE   (OFF_EIMP + NIMP)          // 813001
#define SAMPLE_STRIDE (OFF_FORCE + N_ATOMS*3)  // 819001

#define CHARGE_CONST 18.222615f
#define TH0_SCALE    0.31415927f               // float(pi/10)

typedef float v2f __attribute__((ext_vector_type(2)));
typedef int   v2i __attribute__((ext_vector_type(2)));

// ---- guaranteed hardware global fadd (no return): global_atomic_add_f32 ----
__device__ __forceinline__ void gadd(float* p, float v) {
  asm volatile("global_atomic_add_f32 %0, %1, off"
               :: "v"((unsigned long long)(uintptr_t)p), "v"(v)
               : "memory");
}
// LDS fadd accumulate (ds_add_f32)
__device__ __forceinline__ void ladd(float* p, float v) {
  __hip_atomic_fetch_add(p, v, __ATOMIC_RELAXED, __HIP_MEMORY_SCOPE_WORKGROUP);
}
// gfx1250 async global->LDS copy (128-bit), tracked by ASYNCcnt
__device__ __forceinline__ void async_copy_b128(unsigned lds_byte_addr,
                                                const void* gptr) {
  asm volatile("global_load_async_to_lds_b128 %0, %1, off"
               :: "v"(lds_byte_addr), "v"((unsigned long long)(uintptr_t)gptr)
               : "memory");
}
__device__ __forceinline__ void wait_asynccnt0() {
  asm volatile("s_wait_asynccnt 0x0" ::: "memory");
}

// ---------------- init: zero Force region and E_ub ----------------
__global__ void zero_kernel(float* __restrict__ out) {
  const int per = N_ATOMS * 3 + 1;
  int i = blockIdx.x * blockDim.x + threadIdx.x;
  if (i >= NS * per) return;
  int s = i / per, r = i % per;
  float* base = out + (size_t)s * SAMPLE_STRIDE;
  if (r == 0) base[OFF_EUB] = 0.0f;
  else        base[OFF_FORCE + (r - 1)] = 0.0f;
}

// ---------------- bond ----------------
__global__ void bond_kernel(const float* __restrict__ length_bond,
                            const float* __restrict__ paras_bond,
                            const float* __restrict__ dlength_bond,
                            const int*   __restrict__ bond_index,
                            float* __restrict__ out) {
  int i = blockIdx.x * blockDim.x + threadIdx.x;
  if (i >= NS * NBOND) return;
  int s = i / NBOND, b = i % NBOND;
  float K  = paras_bond[2*b] * 100.0f;
  float r0 = paras_bond[2*b + 1];
  float d  = length_bond[i] - r0;
  float* base = out + (size_t)s * SAMPLE_STRIDE;
  base[OFF_EBOND + b] = K * d * d;
  float c = 2.0f * K * d;
  const float* dl = dlength_bond + (size_t)i * 6;
  int a0 = bond_index[2*b], a1 = bond_index[2*b + 1];
  float* F = base + OFF_FORCE;
  #pragma unroll
  for (int k = 0; k < 3; ++k) gadd(&F[3*a0 + k], dl[k]     * c);
  #pragma unroll
  for (int k = 0; k < 3; ++k) gadd(&F[3*a1 + k], dl[3 + k] * c);
}

// ---------------- angle ----------------
__global__ void angle_kernel(const float* __restrict__ theta_angle,
                             const float* __restrict__ paras_angle,
                             const float* __restrict__ dtheta_angle,
                             const int*   __restrict__ angle_index,
                             float* __restrict__ out) {
  int i = blockIdx.x * blockDim.x + threadIdx.x;
  if (i >= NS * NANG) return;
  int s = i / NANG, a = i % NANG;
  float Ka  = paras_angle[2*a] * 10.0f;
  float th0 = paras_angle[2*a + 1] * TH0_SCALE;
  float d   = theta_angle[i] - th0;
  float* base = out + (size_t)s * SAMPLE_STRIDE;
  base[OFF_EANGLE + a] = Ka * d * d;
  float c = 2.0f * Ka * d;
  const float* dt = dtheta_angle + (size_t)i * 9;
  float* F = base + OFF_FORCE;
  #pragma unroll
  for (int at = 0; at < 3; ++at) {
    int idx = angle_index[3*a + at];
    #pragma unroll
    for (int k = 0; k < 3; ++k) gadd(&F[3*idx + k], dt[3*at + k] * c);
  }
}

// ---------------- torsion ----------------
__global__ void torsion_kernel(const float* __restrict__ sin_cos,     // (NS,NT,8)
                               const float* __restrict__ paras_tor,   // (NT,4)
                               const float* __restrict__ dtheta_tor,  // (NS,NT,12)
                               const int*   __restrict__ tor_index,   // (NT,4)
                               float* __restrict__ out) {
  int i = blockIdx.x * blockDim.x + threadIdx.x;
  if (i >= NS * NTOR) return;
  int s = i / NTOR, t = i % NTOR;
  const float* sc = sin_cos + (size_t)i * 8;
  const float* p  = paras_tor + (size_t)t * 4;
  float E = 0.0f, S = 0.0f;
  #pragma unroll
  for (int n = 0; n < 4; ++n) {
    E += sc[2*n + 1] * p[n];                       // cos terms
    S += sc[2*n]     * p[n] * (float)(n + 1);      // sin * n*coeff
  }
  float* base = out + (size_t)s * SAMPLE_STRIDE;
  base[OFF_ETOR + t] = E;
  float c = -S;
  const float* d = dtheta_tor + (size_t)i * 12;
  float* F = base + OFF_FORCE;
  #pragma unroll
  for (int at = 0; at < 4; ++at) {
    int idx = tor_index[4*t + at];
    #pragma unroll
    for (int k = 0; k < 3; ++k) gadd(&F[3*idx + k], d[3*at + k] * c);
  }
}

// ---------------- imptors ----------------
__global__ void imptors_kernel(const float* __restrict__ cos2,        // (NS,NI)
                               const float* __restrict__ paras_imp,   // (NI,1)
                               const float* __restrict__ dcos2,       // (NS,NI,12)
                               const int*   __restrict__ imp_index,   // (NI,4)
                               float* __restrict__ out) {
  int i = blockIdx.x * blockDim.x + threadIdx.x;
  if (i >= NS * NIMP) return;
  int s = i / NIMP, t = i % NIMP;
  float ki = paras_imp[t];
  float* base = out + (size_t)s * SAMPLE_STRIDE;
  base[OFF_EIMP + t] = ki * (1.0f - cos2[i]);
  float c = -ki;
  const float* d = dcos2 + (size_t)i * 12;
  float* F = base + OFF_FORCE;
  #pragma unroll
  for (int at = 0; at < 4; ++at) {
    int idx = imp_index[4*t + at];
    #pragma unroll
    for (int k = 0; k < 3; ++k) gadd(&F[3*idx + k], d[3*at + k] * c);
  }
}

// ---------------- nonbonded (vdW + charge): the bandwidth hot spot ----------------
#define NBT 512
#define PPB 16384

__global__ __launch_bounds__(NBT)
void nb_kernel(const float* __restrict__ length_vdw,    // (NS,NV)
               const float* __restrict__ vdw14,         // (NV)
               const float* __restrict__ charge14,      // (NV)
               const float* __restrict__ paras_vdw,     // (N_ATOMS,2)
               const float* __restrict__ paras_charge,  // (N_ATOMS)
               const float* __restrict__ dlength_vdw,   // (NS,NV,6)
               const int*   __restrict__ nonbonded,     // (2,NV)
               const int*   __restrict__ nb_index,      // (NV,2)
               float* __restrict__ out) {
  __shared__ __align__(16) float fbuf[N_ATOMS * 3]; // 24 KB per-sample force accum
  __shared__ __align__(16) float pv[N_ATOMS * 2];   // 16 KB raw vdW params
  __shared__ __align__(16) float pc[N_ATOMS];       //  8 KB raw charges

  const int s = blockIdx.y;
  for (int k = threadIdx.x; k < N_ATOMS * 3; k += NBT) fbuf[k] = 0.0f;

  // ---- gfx1250 async global->LDS staging of the parameter tables ----
  {
    const unsigned lds_pv = (unsigned)(uintptr_t)&pv[0];   // LDS byte offset (addr[31:0])
    const unsigned lds_pc = (unsigned)(uintptr_t)&pc[0];
    // pv: 16000 B = 1000 x b128
    for (int k = threadIdx.x; k < (N_ATOMS * 2) / 4; k += NBT)
      async_copy_b128(lds_pv + 16u * k, (const char*)paras_vdw + 16u * k);
    // pc: 8000 B = 500 x b128
    for (int k = threadIdx.x; k < N_ATOMS / 4; k += NBT)
      async_copy_b128(lds_pc + 16u * k, (const char*)paras_charge + 16u * k);
    wait_asynccnt0();
  }
  __syncthreads();

  const int v0 = blockIdx.x * PPB;
  const int v1 = (v0 + PPB < NVDW) ? (v0 + PPB) : NVDW;

  const float* __restrict__ rr = length_vdw  + (size_t)s * NVDW;
  const float* __restrict__ dl = dlength_vdw + (size_t)s * NVDW * 6;
  float* base = out + (size_t)s * SAMPLE_STRIDE;
  float* __restrict__ evdw = base + OFF_EVDW;
  float* __restrict__ ech  = base + OFF_ECH;

  const float K2 = (CHARGE_CONST * 0.1f) * (CHARGE_CONST * 0.1f); // (CHARGE/10)^2

  for (int v = v0 + (int)threadIdx.x; v < v1; v += NBT) {
    const int i = nonbonded[v];
    const int j = nonbonded[NVDW + v];
    const float sig = pv[2*i] + pv[2*j];
    const float eps = pv[2*i + 1] * 0.1f * (pv[2*j + 1] * 0.1f) * vdw14[v];
    const float cc  = K2 * pc[i] * pc[j] * charge14[v];

    const float r = __builtin_nontemporal_load(rr + v);

    // one precise reciprocal; everything else is multiplies
    const float ir  = 1.0f / r;
    const float ir2 = ir * ir;
    const float ir6 = ir2 * ir2 * ir2;
    const float sig2 = sig * sig;
    const float sig6 = sig2 * sig2 * sig2;
    const float t  = sig6 * ir6;

    __builtin_nontemporal_store(eps * (t * t - 2.0f * t), evdw + v);
    __builtin_nontemporal_store(cc * ir, ech + v);

    const float fc = 12.0f * eps * (t * (1.0f - t)) * ir - cc * ir2;

    // 24 B of derivatives, 8-byte aligned -> three NT b64 loads
    const v2f* d6 = (const v2f*)(dl + (size_t)v * 6);
    const v2f d01 = __builtin_nontemporal_load(d6 + 0);
    const v2f d23 = __builtin_nontemporal_load(d6 + 1);
    const v2f d45 = __builtin_nontemporal_load(d6 + 2);

    const v2i aa = *(const v2i*)(nb_index + 2 * (size_t)v);
    const int a0 = aa.x, a1 = aa.y;
    ladd(&fbuf[3*a0 + 0], d01.x * fc);
    ladd(&fbuf[3*a0 + 1], d01.y * fc);
    ladd(&fbuf[3*a0 + 2], d23.x * fc);
    ladd(&fbuf[3*a1 + 0], d23.y * fc);
    ladd(&fbuf[3*a1 + 1], d45.x * fc);
    ladd(&fbuf[3*a1 + 2], d45.y * fc);
  }
  __syncthreads();

  float* __restrict__ Fg = base + OFF_FORCE;
  for (int k = threadIdx.x; k < N_ATOMS * 3; k += NBT) gadd(&Fg[k], fbuf[k]);
}

// ---------------- host launch ----------------
extern "C" void kernel_launch(void* const* d_in, const int* in_sizes, int n_in,
                              void* d_out, int out_size, void* d_ws, size_t ws_size,
                              hipStream_t stream) {
  (void)in_sizes; (void)n_in; (void)out_size; (void)d_ws; (void)ws_size;

  const float* length_bond     = (const float*)d_in[0];
  const float* theta_angle     = (const float*)d_in[1];
  const float* length_vdw      = (const float*)d_in[2];
  const float* sin_cos_torsion = (const float*)d_in[3];
  const float* cos2_imptors    = (const float*)d_in[4];
  const float* vdw14           = (const float*)d_in[5];
  const float* charge14        = (const float*)d_in[6];
  const float* paras_bond      = (const float*)d_in[7];
  const float* paras_angle     = (const float*)d_in[8];
  const float* paras_vdw       = (const float*)d_in[9];
  const float* paras_charge    = (const float*)d_in[10];
  const float* paras_torsion   = (const float*)d_in[11];
  const float* paras_imptors   = (const float*)d_in[12];
  const float* dlength_bond    = (const float*)d_in[13];
  const float* dtheta_angle    = (const float*)d_in[14];
  const float* dlength_vdw     = (const float*)d_in[15];
  const float* dtheta_torsion  = (const float*)d_in[16];
  const float* dcos2_imptors   = (const float*)d_in[17];
  const int*   nonbonded       = (const int*)d_in[18];
  const int*   bond_index      = (const int*)d_in[19];
  const int*   angle_index     = (const int*)d_in[20];
  const int*   nonbonded_index = (const int*)d_in[21];
  const int*   torsion_index   = (const int*)d_in[22];
  const int*   imptors_index   = (const int*)d_in[23];
  float* out = (float*)d_out;

  // 1) zero Force + E_ub (must precede all atomic accumulation; same stream -> ordered)
  {
    int total = NS * (N_ATOMS * 3 + 1);
    zero_kernel<<<(total + 255) / 256, 256, 0, stream>>>(out);
  }
  // 2) small term families (direct hardware global fadd atomics)
  bond_kernel<<<(NS * NBOND + 255) / 256, 256, 0, stream>>>(
      length_bond, paras_bond, dlength_bond, bond_index, out);
  angle_kernel<<<(NS * NANG + 255) / 256, 256, 0, stream>>>(
      theta_angle, paras_angle, dtheta_angle, angle_index, out);
  torsion_kernel<<<(NS * NTOR + 255) / 256, 256, 0, stream>>>(
      sin_cos_torsion, paras_torsion, dtheta_torsion, torsion_index, out);
  imptors_kernel<<<(NS * NIMP + 255) / 256, 256, 0, stream>>>(
      cos2_imptors, paras_imptors, dcos2_imptors, imptors_index, out);
  // 3) nonbonded hot spot: async-LDS staged tables, LDS-accumulated scatter,
  //    NT-hinted streaming of the 154 MB derivative tensor
  {
    dim3 grid((NVDW + PPB - 1) / PPB, NS);
    nb_kernel<<<grid, NBT, 0, stream>>>(
        length_vdw, vdw14, charge14, paras_vdw, paras_charge,
        dlength_vdw, nonbonded, nonbonded_index, out);
  }
}